// MultiSAGE_14542759264834
// MI455X (gfx1250) — compile-verified
//
#include <hip/hip_runtime.h>
#include <math.h>

typedef float v2f __attribute__((ext_vector_type(2)));
typedef float v8f __attribute__((ext_vector_type(8)));
typedef int   v2i __attribute__((ext_vector_type(2)));

static constexpr int   Nn   = 100000;
static constexpr int   Dd   = 128;
static constexpr long  NDl  = (long)Nn * Dd;       // 12,800,000 floats
static constexpr int   TILES = Nn / 16;            // 6250
static constexpr float EPSf = 1e-5f;

// ---- workspace layout (float offsets) ----
static constexpr long AGG0 = 0;
static constexpr long AGG1 = NDl;
static constexpr long AGG2 = 2 * NDl;
static constexpr long AGGF = 3 * NDl;
static constexpr long COMB = 4 * NDl;
static constexpr long CNT0 = 5 * NDl;
static constexpr long CNT1 = CNT0 + Nn;
static constexpr long CNT2 = CNT0 + 2L * Nn;
static constexpr long CNTF = CNT0 + 3L * Nn;
static constexpr long SUMO = CNT0 + 4L * Nn;       // 128 col sums
static constexpr long SQO  = SUMO + 128;           // 128 col sum-of-squares
static constexpr long WEFF = SUMO + 256;           // 128x128 fused right-weight
static constexpr long BEFF = WEFF + 16384;         // 128 fused bias

// ---- gfx1250 async LDS copy availability ----
#if defined(__has_builtin)
#  if __has_builtin(__builtin_amdgcn_global_load_async_to_lds_b64)
#    define ASYNC_LDS 1
#  endif
#  if __has_builtin(__builtin_amdgcn_s_wait_asynccnt)
#    define HAVE_WAIT_ASYNC 1
#  endif
#endif

#define AS1P __attribute__((address_space(1)))
#define AS3P __attribute__((address_space(3)))

// -------------------------------------------------- zero (float4 stores)
__global__ void k_zero(float* __restrict__ p, long n4) {
    long i = (long)blockIdx.x * blockDim.x + threadIdx.x;
    if (i < n4) *(float4*)(p + 4 * i) = make_float4(0.f, 0.f, 0.f, 0.f);
}

// -------------------------------------------------- Weff = sum wt_i*Wr_i + Wra ; beff likewise
__global__ void k_prep(const float* __restrict__ Wr0, const float* __restrict__ Wr1,
                       const float* __restrict__ Wr2, const float* __restrict__ Wra,
                       const float* __restrict__ bl0, const float* __restrict__ bl1,
                       const float* __restrict__ bl2, const float* __restrict__ bla,
                       const float* __restrict__ wt,
                       float* __restrict__ weff, float* __restrict__ beff) {
    int i = blockIdx.x * blockDim.x + threadIdx.x;
    float w0 = wt[0], w1 = wt[1], w2 = wt[2];
    if (i < Dd * Dd) weff[i] = w0 * Wr0[i] + w1 * Wr1[i] + w2 * Wr2[i] + Wra[i];
    if (i < Dd)      beff[i] = w0 * bl0[i] + w1 * bl1[i] + w2 * bl2[i] + bla[i];
}

// -------------------------------------------------- edge scatter-add (1 wave / edge, coalesced)
__global__ void k_scatter(const int* __restrict__ e, int E,
                          const float* __restrict__ x,
                          float* __restrict__ agg, float* __restrict__ cnt) {
    int lane = threadIdx.x & 31;
    int edge = (int)(((long)blockIdx.x * blockDim.x + threadIdx.x) >> 5);
    if (edge >= E) return;
    int s = e[edge];
    int d = e[E + edge];
    const float* xs = x + (long)s * Dd;
    float*       ad = agg + (long)d * Dd;
#pragma unroll
    for (int c = 0; c < 4; ++c)
        atomicAdd(ad + lane + 32 * c, xs[lane + 32 * c]);
    if (lane == 0) atomicAdd(cnt + d, 1.0f);
}

// -------------------------------------------------- weight staging into LDS (fragment-swizzled)
// LDS layout: lB[((k4*8 + j)*64) + lane*2 + {0,1}] = W[(j*16+laneM)*128 + k4*4 + koff + {0,1}]
// -> compute-side ds reads touch 64 consecutive dwords per wave: all 64 banks, conflict-free.
__device__ __forceinline__ void stage_B(const float* __restrict__ W, float* lB,
                                        int waveId, int lane, int laneM, int koff) {
#pragma unroll
    for (int kk = 0; kk < 4; ++kk) {
        int k4 = waveId * 4 + kk;               // 8 waves x 4 = all 32 k-steps
#pragma unroll
        for (int j = 0; j < 8; ++j) {
            const float* g  = W + (j * 16 + laneM) * Dd + k4 * 4 + koff;
            float*       l  = lB + ((k4 * 8 + j) * 64) + lane * 2;
#if defined(ASYNC_LDS)
            __builtin_amdgcn_global_load_async_to_lds_b64(
                (AS1P v2i*)g, (AS3P v2i*)l, 0, 0);
#else
            *(v2f*)l = *(const v2f*)g;
#endif
        }
    }
}

__device__ __forceinline__ void stage_sync() {
#if defined(ASYNC_LDS)
#  if defined(HAVE_WAIT_ASYNC)
    __builtin_amdgcn_s_wait_asynccnt(0);
#  else
    asm volatile("s_wait_asynccnt 0x0" ::: "memory");
#  endif
#endif
    __syncthreads();
}

// -------------------------------------------------- fp32 WMMA K-sweep, B from LDS
// A-frag (16x4 f32): lanes 0-15 -> K={0,1} of row lane; lanes 16-31 -> K={2,3} of row lane-16.
__device__ __forceinline__ void mma_strip_lds(const float* __restrict__ Arow, float scale,
                                              const float* lB, int lane, int koff, v8f* acc) {
    for (int k0 = 0; k0 < Dd; k0 += 4) {
        v2f a = *(const v2f*)(Arow + k0 + koff);
        a *= scale;
        const float* lb = lB + (k0 >> 2) * 512 + lane * 2;
#pragma unroll
        for (int j = 0; j < 8; ++j) {
            v2f b = *(const v2f*)(lb + j * 64);
            acc[j] = __builtin_amdgcn_wmma_f32_16x16x4_f32(
                false, a, false, b, (short)0, acc[j], false, false);
        }
    }
}

// three-way summed A (mean over concatenated edge set: agg_all = agg0+agg1+agg2)
__device__ __forceinline__ void mma_strip3_lds(const float* __restrict__ A0,
                                               const float* __restrict__ A1,
                                               const float* __restrict__ A2, float scale,
                                               const float* lB, int lane, int koff, v8f* acc) {
    for (int k0 = 0; k0 < Dd; k0 += 4) {
        v2f a = *(const v2f*)(A0 + k0 + koff);
        a += *(const v2f*)(A1 + k0 + koff);
        a += *(const v2f*)(A2 + k0 + koff);
        a *= scale;
        const float* lb = lB + (k0 >> 2) * 512 + lane * 2;
#pragma unroll
        for (int j = 0; j < 8; ++j) {
            v2f b = *(const v2f*)(lb + j * 64);
            acc[j] = __builtin_amdgcn_wmma_f32_16x16x4_f32(
                false, a, false, b, (short)0, acc[j], false, false);
        }
    }
}

// -------------------------------------------------- fused combine: 5 GEMMs + ReLU + BN stats
__global__ __launch_bounds__(256) void k_gemm_combine(
    const float* __restrict__ feat,
    const float* __restrict__ agg0, const float* __restrict__ agg1, const float* __restrict__ agg2,
    const float* __restrict__ cnt0, const float* __restrict__ cnt1, const float* __restrict__ cnt2,
    const float* __restrict__ Wl0, const float* __restrict__ Wl1, const float* __restrict__ Wl2,
    const float* __restrict__ Wla,
    const float* __restrict__ weff, const float* __restrict__ beff,
    const float* __restrict__ wt,
    float* __restrict__ comb, float* __restrict__ sum, float* __restrict__ sumsq) {
    __shared__ float lB[16384];                     // one 128x128 weight, fragment-swizzled

    int  lane   = threadIdx.x & 31;
    int  waveId = threadIdx.x >> 5;
    int  tile   = blockIdx.x * 8 + waveId;
    bool active = tile < TILES;                     // wave-uniform
    int  laneM   = lane & 15;
    int  koff    = (lane & 16) >> 3;                // 0 or 2
    int  rowHalf = (lane & 16) >> 1;                // 0 or 8
    long r       = (long)(active ? tile : 0) * 16 + laneM;

    v8f acc[8];
#pragma unroll
    for (int j = 0; j < 8; ++j) {
        float bv = beff[j * 16 + laneM];
        acc[j] = (v8f){bv, bv, bv, bv, bv, bv, bv, bv};
    }

    float c0 = cnt0[r], c1 = cnt1[r], c2 = cnt2[r];
    float w0 = wt[0], w1 = wt[1], w2 = wt[2];

    // --- source 0: wt0 * mean0 @ Wl0^T ---
    stage_B(Wl0, lB, waveId, lane, laneM, koff);
    stage_sync();
    if (active) mma_strip_lds(agg0 + r * Dd, w0 / fmaxf(c0, 1.f), lB, lane, koff, acc);
    __syncthreads();
    // --- source 1 ---
    stage_B(Wl1, lB, waveId, lane, laneM, koff);
    stage_sync();
    if (active) mma_strip_lds(agg1 + r * Dd, w1 / fmaxf(c1, 1.f), lB, lane, koff, acc);
    __syncthreads();
    // --- source 2 ---
    stage_B(Wl2, lB, waveId, lane, laneM, koff);
    stage_sync();
    if (active) mma_strip_lds(agg2 + r * Dd, w2 / fmaxf(c2, 1.f), lB, lane, koff, acc);
    __syncthreads();
    // --- source 3: mean over concatenated edges @ Wla^T (agg_all = agg0+agg1+agg2) ---
    stage_B(Wla, lB, waveId, lane, laneM, koff);
    stage_sync();
    if (active) mma_strip3_lds(agg0 + r * Dd, agg1 + r * Dd, agg2 + r * Dd,
                               1.f / fmaxf(c0 + c1 + c2, 1.f), lB, lane, koff, acc);
    __syncthreads();
    // --- source 4: feat @ Weff^T (all four self-terms folded) ---
    stage_B(weff, lB, waveId, lane, laneM, koff);
    stage_sync();
    if (active) mma_strip_lds(feat + r * Dd, 1.f, lB, lane, koff, acc);

    if (active) {
#pragma unroll
        for (int j = 0; j < 8; ++j) {
            int   col = j * 16 + laneM;
            float s1 = 0.f, s2 = 0.f;
#pragma unroll
            for (int v = 0; v < 8; ++v) {
                float x = fmaxf(acc[j][v], 0.f);    // ReLU
                comb[((long)tile * 16 + rowHalf + v) * Dd + col] = x;
                s1 += x;
                s2 += x * x;
            }
            atomicAdd(sum + col, s1);               // BN batch statistics
            atomicAdd(sumsq + col, s2);
        }
    }
}

// -------------------------------------------------- BN affine, in place on comb
__global__ void k_bn(float* __restrict__ comb,
                     const float* __restrict__ sum, const float* __restrict__ sumsq,
                     const float* __restrict__ gamma, const float* __restrict__ beta) {
    long i = ((long)blockIdx.x * blockDim.x + threadIdx.x) * 4;
    if (i >= NDl) return;
    int d = (int)(i & (Dd - 1));
    const float invN = 1.0f / (float)Nn;
    float4 v = *(float4*)(comb + i);
    float o[4] = {v.x, v.y, v.z, v.w};
#pragma unroll
    for (int c = 0; c < 4; ++c) {
        int   dd  = d + c;
        float mu  = sum[dd] * invN;
        float var = sumsq[dd] * invN - mu * mu;
        float a   = gamma[dd] * rsqrtf(var + EPSf);
        o[c]      = (o[c] - mu) * a + beta[dd];
    }
    *(float4*)(comb + i) = make_float4(o[0], o[1], o[2], o[3]);
}

// -------------------------------------------------- final conv: out = meanF@Wf^T + bf + comb@Wrf^T
__global__ __launch_bounds__(256) void k_gemm_final(
    const float* __restrict__ comb,
    const float* __restrict__ aggF, const float* __restrict__ cntF,
    const float* __restrict__ Wf, const float* __restrict__ bf,
    const float* __restrict__ Wrf, float* __restrict__ out) {
    __shared__ float lB[16384];

    int  lane   = threadIdx.x & 31;
    int  waveId = threadIdx.x >> 5;
    int  tile   = blockIdx.x * 8 + waveId;
    bool active = tile < TILES;
    int  laneM   = lane & 15;
    int  koff    = (lane & 16) >> 3;
    int  rowHalf = (lane & 16) >> 1;
    long r       = (long)(active ? tile : 0) * 16 + laneM;

    v8f acc[8];
#pragma unroll
    for (int j = 0; j < 8; ++j) {
        float bv = bf[j * 16 + laneM];
        acc[j] = (v8f){bv, bv, bv, bv, bv, bv, bv, bv};
    }

    float cF = cntF[r];

    stage_B(Wf, lB, waveId, lane, laneM, koff);
    stage_sync();
    if (active) mma_strip_lds(aggF + r * Dd, 1.f / fmaxf(cF, 1.f), lB, lane, koff, acc);
    __syncthreads();
    stage_B(Wrf, lB, waveId, lane, laneM, koff);
    stage_sync();
    if (active) mma_strip_lds(comb + r * Dd, 1.f, lB, lane, koff, acc);

    if (active) {
#pragma unroll
        for (int j = 0; j < 8; ++j) {
            int col = j * 16 + laneM;
#pragma unroll
            for (int v = 0; v < 8; ++v)
                out[((long)tile * 16 + rowHalf + v) * Dd + col] = acc[j][v];
        }
    }
}

// ==================================================================
extern "C" void kernel_launch(void* const* d_in, const int* in_sizes, int n_in,
                              void* d_out, int out_size, void* d_ws, size_t ws_size,
                              hipStream_t stream) {
    const float* feat  = (const float*)d_in[0];
    const int*   e0    = (const int*)d_in[1];
    const int*   e1    = (const int*)d_in[2];
    const int*   e2    = (const int*)d_in[3];
    const float* Wl0   = (const float*)d_in[4];
    const float* bl0   = (const float*)d_in[5];
    const float* Wr0   = (const float*)d_in[6];
    const float* Wl1   = (const float*)d_in[7];
    const float* bl1   = (const float*)d_in[8];
    const float* Wr1   = (const float*)d_in[9];
    const float* Wl2   = (const float*)d_in[10];
    const float* bl2   = (const float*)d_in[11];
    const float* Wr2   = (const float*)d_in[12];
    const float* Wla   = (const float*)d_in[13];
    const float* bla   = (const float*)d_in[14];
    const float* Wra   = (const float*)d_in[15];
    const float* Wf    = (const float*)d_in[16];
    const float* bf    = (const float*)d_in[17];
    const float* Wrf   = (const float*)d_in[18];
    const float* gamma = (const float*)d_in[19];
    const float* beta  = (const float*)d_in[20];
    const float* wt    = (const float*)d_in[21];

    float* ws   = (float*)d_ws;
    float* outp = (float*)d_out;

    const int E0 = in_sizes[1] / 2;
    const int E1 = in_sizes[2] / 2;
    const int E2 = in_sizes[3] / 2;

    // 1) zero aggregation buffers, counts, BN stats (ws is poisoned)
    {
        long n4 = (4 * NDl) / 4;
        k_zero<<<(int)((n4 + 255) / 256), 256, 0, stream>>>(ws + AGG0, n4);
        long n4b = (4L * Nn + 256) / 4;
        k_zero<<<(int)((n4b + 255) / 256), 256, 0, stream>>>(ws + CNT0, n4b);
    }

    // 2) fused right-weight / bias
    k_prep<<<(Dd * Dd + 255) / 256, 256, 0, stream>>>(Wr0, Wr1, Wr2, Wra,
                                                      bl0, bl1, bl2, bla, wt,
                                                      ws + WEFF, ws + BEFF);

    // 3) per-etype scatter aggregation of features
    {
        k_scatter<<<(E0 * 32 + 255) / 256, 256, 0, stream>>>(e0, E0, feat, ws + AGG0, ws + CNT0);
        k_scatter<<<(E1 * 32 + 255) / 256, 256, 0, stream>>>(e1, E1, feat, ws + AGG1, ws + CNT1);
        k_scatter<<<(E2 * 32 + 255) / 256, 256, 0, stream>>>(e2, E2, feat, ws + AGG2, ws + CNT2);
    }

    // 4) fused combine GEMMs + ReLU + BN stats
    {
        int blocks = (TILES + 7) / 8;   // 8 waves (tiles) per 256-thread block
        k_gemm_combine<<<blocks, 256, 0, stream>>>(
            feat, ws + AGG0, ws + AGG1, ws + AGG2,
            ws + CNT0, ws + CNT1, ws + CNT2,
            Wl0, Wl1, Wl2, Wla, ws + WEFF, ws + BEFF, wt,
            ws + COMB, ws + SUMO, ws + SQO);
    }

    // 5) BatchNorm affine in place
    k_bn<<<(int)((NDl / 4 + 255) / 256), 256, 0, stream>>>(ws + COMB, ws + SUMO, ws + SQO,
                                                           gamma, beta);

    // 6) scatter-aggregate normalized features over the concatenated edge set
    {
        k_scatter<<<(E0 * 32 + 255) / 256, 256, 0, stream>>>(e0, E0, ws + COMB, ws + AGGF, ws + CNTF);
        k_scatter<<<(E1 * 32 + 255) / 256, 256, 0, stream>>>(e1, E1, ws + COMB, ws + AGGF, ws + CNTF);
        k_scatter<<<(E2 * 32 + 255) / 256, 256, 0, stream>>>(e2, E2, ws + COMB, ws + AGGF, ws + CNTF);
    }

    // 7) final SAGE conv -> d_out
    {
        int blocks = (TILES + 7) / 8;
        k_gemm_final<<<blocks, 256, 0, stream>>>(ws + COMB, ws + AGGF, ws + CNTF,
                                                 Wf, bf, Wrf, outp);
    }
}